// SAGE_18468359373225
// MI455X (gfx1250) — compile-verified
//
#include <hip/hip_runtime.h>

#define NN 50000
#define NE 800000
#define DIM 128
#define LDS_PITCH 132   // 128 + 4 pad: 16B-aligned rows, conflict-free (4m+k)%64 bank map

typedef float v2f __attribute__((ext_vector_type(2)));
typedef float v8f __attribute__((ext_vector_type(8)));

__device__ __forceinline__ void atomic_add_f32(float* p, float v) {
  // relaxed, device-scope -> global_atomic_add_f32 on gfx1250
  __hip_atomic_fetch_add(p, v, __ATOMIC_RELAXED, __HIP_MEMORY_SCOPE_AGENT);
}

__global__ void zero_f32(float* __restrict__ p, int n) {
  int i = blockIdx.x * blockDim.x + threadIdx.x;
  if (i < n) p[i] = 0.0f;
}

// One 32-lane group per edge: float4 gather of feat[src], 4 atomic adds into agg[dst].
__global__ void sage_scatter(const float* __restrict__ feat,
                             const int* __restrict__ src,
                             const int* __restrict__ dst,
                             float* __restrict__ agg,
                             float* __restrict__ cnt,
                             int do_count) {
  int t = blockIdx.x * blockDim.x + threadIdx.x;
  int e = t >> 5;
  int lane = t & 31;
  if (e >= NE) return;
  int s = src[e];
  int d = dst[e];
  const float4 v = *(const float4*)(feat + (size_t)s * DIM + lane * 4);
  float* o = agg + (size_t)d * DIM + lane * 4;
  atomic_add_f32(o + 0, v.x);
  atomic_add_f32(o + 1, v.y);
  atomic_add_f32(o + 2, v.z);
  atomic_add_f32(o + 3, v.w);
  if (do_count && lane == 0) atomic_add_f32(cnt + d, 1.0f);
}

// out[n,o] = inv_cnt[n] * (agg[n,:] @ Wl[o,:]) + (xin[n,:] @ Wr[o,:]) + bias[o], optional ReLU.
// Block = 256 threads = 8 waves. blockIdx.x -> 16 nodes; wave -> 16 output features.
// A tiles (agg + xin, 16x128 each) staged once per block into LDS with coalesced
// float4 loads; WMMA A-fragments then come from conflict-free ds_load_b64.
// Per wave: 16x16 C tile, K=128 swept in chunks of 4 with V_WMMA_F32_16X16X4_F32.
__global__ void __launch_bounds__(256)
sage_combine(const float* __restrict__ agg,
             const float* __restrict__ cnt,
             const float* __restrict__ xin,
             const float* __restrict__ Wl,   // [DIM out, DIM in] row-major
             const float* __restrict__ Wr,   // [DIM out, DIM in] row-major
             const float* __restrict__ bias, // [DIM]
             float* __restrict__ out,        // [NN, DIM]
             int relu) {
  __shared__ float sA[16 * LDS_PITCH];      // agg tile
  __shared__ float sX[16 * LDS_PITCH];      // xin tile

  const int tile = blockIdx.x;              // node tile: 16 rows
  const int tid  = threadIdx.x;
  const int wave = tid >> 5;                // 0..7 -> feature tile
  const int lane = tid & 31;
  const int m    = lane & 15;               // A row within tile; also C/D column
  const int half = lane >> 4;               // 0: K+{0,1}, 1: K+{2,3} (ISA 16x4 f32 layout)
  const int koff = half << 1;

  // ---- cooperative tile staging: 16 rows x 128 cols = 512 float4s, 2 per thread ----
#pragma unroll
  for (int i = tid; i < 512; i += 256) {
    const int r = i >> 5;                   // 32 float4s per row
    const int c = (i & 31) << 2;
    const size_t g = (size_t)(tile * 16 + r) * DIM + c;
    *(float4*)&sA[r * LDS_PITCH + c] = *(const float4*)(agg + g);
    *(float4*)&sX[r * LDS_PITCH + c] = *(const float4*)(xin + g);
  }
  __syncthreads();

  const int feat = wave * 16 + m;           // output feature (B column / C column)
  const float* bL = Wl + (size_t)feat * DIM;  // B[k][n] = Wl[feat][k]
  const float* bR = Wr + (size_t)feat * DIM;
  const float* aL = &sA[m * LDS_PITCH];
  const float* aR = &sX[m * LDS_PITCH];

  v8f accL = {};
  v8f accR = {};

#pragma unroll
  for (int kc = 0; kc < DIM; kc += 4) {
    const int k = kc + koff;
    v2f a0 = *(const v2f*)(aL + k);         // ds_load_b64, conflict-free
    v2f b0 = *(const v2f*)(bL + k);         // global, L1/L2-resident weights
    v2f a1 = *(const v2f*)(aR + k);
    v2f b1 = *(const v2f*)(bR + k);
    // (neg_a, A, neg_b, B, c_mod, C, reuse_a, reuse_b)
    accL = __builtin_amdgcn_wmma_f32_16x16x4_f32(false, a0, false, b0,
                                                 (short)0, accL, false, false);
    accR = __builtin_amdgcn_wmma_f32_16x16x4_f32(false, a1, false, b1,
                                                 (short)0, accR, false, false);
  }

  const float bcol = bias[feat];
#pragma unroll
  for (int v = 0; v < 8; ++v) {
    // C/D layout: lanes 0-15 hold rows v, lanes 16-31 hold rows v+8; column = lane&15
    const int row  = v + half * 8;
    const int node = tile * 16 + row;
    const float inv = 1.0f / fmaxf(cnt[node], 1.0f);
    float o = accL[v] * inv + accR[v] + bcol;
    if (relu) o = fmaxf(o, 0.0f);
    out[(size_t)node * DIM + feat] = o;
  }
}

extern "C" void kernel_launch(void* const* d_in, const int* in_sizes, int n_in,
                              void* d_out, int out_size, void* d_ws, size_t ws_size,
                              hipStream_t stream) {
  (void)in_sizes; (void)n_in; (void)out_size; (void)ws_size;

  const float* x   = (const float*)d_in[0];
  const int*  edge = (const int*)d_in[1];   // [2, NE]: row 0 = src, row 1 = dst
  const int*  src  = edge;
  const int*  dst  = edge + NE;
  const float* Wl0 = (const float*)d_in[2];
  const float* bl0 = (const float*)d_in[3];
  const float* Wr0 = (const float*)d_in[4];
  const float* Wl1 = (const float*)d_in[5];
  const float* bl1 = (const float*)d_in[6];
  const float* Wr1 = (const float*)d_in[7];
  float* out = (float*)d_out;

  // Workspace layout: h [NN*DIM] | agg [NN*DIM] | cnt [NN]
  float* h   = (float*)d_ws;
  float* agg = h + (size_t)NN * DIM;
  float* cnt = agg + (size_t)NN * DIM;

  const int scatter_threads = NE * 32;

  // ---- Layer 0 ----
  {
    int nz = NN * DIM + NN;               // agg and cnt are contiguous
    zero_f32<<<(nz + 255) / 256, 256, 0, stream>>>(agg, nz);
  }
  sage_scatter<<<(scatter_threads + 255) / 256, 256, 0, stream>>>(
      x, src, dst, agg, cnt, /*do_count=*/1);
  sage_combine<<<NN / 16, 256, 0, stream>>>(agg, cnt, x, Wl0, Wr0, bl0, h, /*relu=*/1);

  // ---- Layer 1 (cnt reused: same graph) ----
  {
    int nz = NN * DIM;
    zero_f32<<<(nz + 255) / 256, 256, 0, stream>>>(agg, nz);
  }
  sage_scatter<<<(scatter_threads + 255) / 256, 256, 0, stream>>>(
      h, src, dst, agg, cnt, /*do_count=*/0);
  sage_combine<<<NN / 16, 256, 0, stream>>>(agg, cnt, h, Wl1, Wr1, bl1, out, /*relu=*/0);
}